// WenAttn_42657615184437
// MI455X (gfx1250) — compile-verified
//
#include <hip/hip_runtime.h>

typedef __attribute__((ext_vector_type(2))) float v2f;
typedef __attribute__((ext_vector_type(8))) float v8f;

#define H 1024
#define S 65536

// Workspace layout (float offsets)
#define WS_V     0        // v = W^T hidden           [1024]
#define WS_BMAX  1024     // per-block score maxima   [512]
#define WS_GMAX  1536     // global max               [1]
#define WS_BSUM  1600     // per-block exp sums       [256]
#define WS_GSUM  1856     // global sum               [1]

// ---------------------------------------------------------------------------
// Kernel 1: v[j] = sum_i W[i][j] * hidden[i]   (4 MB stream, coalesced over j)
// ---------------------------------------------------------------------------
__global__ void k_compute_v(const float* __restrict__ W,
                            const float* __restrict__ hidden,
                            float* __restrict__ v) {
    int j = blockIdx.x * 256 + threadIdx.x;
    float acc = 0.0f;
#pragma unroll 4
    for (int i = 0; i < H; ++i)
        acc = fmaf(W[(size_t)i * H + j], hidden[i], acc);
    v[j] = acc;
}

// ---------------------------------------------------------------------------
// Kernel 2: scores[s] = enc[s,:] . v  via V_WMMA_F32_16X16X4_F32
// One wave computes 16 rows; block (8 waves) computes 128 rows; 512 blocks.
// A: lane(l<16) holds enc[row][k+0..1], lane(l>=16) holds enc[row][k+2..3]
// B: v[k..k+3] replicated across N (same per-lane 8B slice of v)
// D: every N column equals the row score; lanes 0/16 write 8 rows each.
// Also produces per-block max for the softmax.
// ---------------------------------------------------------------------------
__global__ void k_scores_wmma(const float* __restrict__ enc,
                              const float* __restrict__ v,
                              float* __restrict__ scores,
                              float* __restrict__ bmax) {
    __shared__ float smax[8];

    const int lane = threadIdx.x & 31;
    const int wave = threadIdx.x >> 5;
    const int half = lane >> 4;      // 0: K lo pair, 1: K hi pair
    const int m    = lane & 15;      // row within tile

    const size_t row  = (size_t)blockIdx.x * 128 + wave * 16 + m;
    const float* arow = enc + row * H + half * 2;
    const float* vk   = v + half * 2;

    v8f acc = {};
#pragma unroll 8
    for (int k = 0; k < H; k += 4) {
        v2f a = *(const v2f*)(arow + k);   // 8B of this row
        v2f b = *(const v2f*)(vk + k);     // 8B of v (L0/L2 resident)
        acc = __builtin_amdgcn_wmma_f32_16x16x4_f32(
            /*neg_a=*/false, a, /*neg_b=*/false, b,
            /*c_mod=*/(short)0, acc, /*reuse_a=*/false, /*reuse_b=*/false);
    }

    // lane 0 holds rows base..base+7 in acc[0..7]; lane 16 holds base+8..+15
    if (m == 0) {
        float* dst = scores + (size_t)blockIdx.x * 128 + wave * 16 + half * 8;
#pragma unroll
        for (int r = 0; r < 8; ++r) dst[r] = acc[r];
    }

    // block max (every lane has a replica of 8 of the wave's 16 row scores)
    float lm = acc[0];
#pragma unroll
    for (int r = 1; r < 8; ++r) lm = fmaxf(lm, acc[r]);
#pragma unroll
    for (int off = 16; off > 0; off >>= 1)
        lm = fmaxf(lm, __shfl_xor(lm, off, 32));
    if (lane == 0) smax[wave] = lm;
    __syncthreads();
    if (threadIdx.x == 0) {
        float bm = smax[0];
#pragma unroll
        for (int w = 1; w < 8; ++w) bm = fmaxf(bm, smax[w]);
        bmax[blockIdx.x] = bm;
    }
}

// ---------------------------------------------------------------------------
// Kernel 3: reduce 512 block maxima -> global max (single block, deterministic)
// ---------------------------------------------------------------------------
__global__ void k_reduce_max(const float* __restrict__ bmax,
                             float* __restrict__ gmax) {
    __shared__ float sm[256];
    sm[threadIdx.x] = fmaxf(bmax[threadIdx.x], bmax[threadIdx.x + 256]);
    __syncthreads();
    for (int off = 128; off > 0; off >>= 1) {
        if ((int)threadIdx.x < off)
            sm[threadIdx.x] = fmaxf(sm[threadIdx.x], sm[threadIdx.x + off]);
        __syncthreads();
    }
    if (threadIdx.x == 0) *gmax = sm[0];
}

// ---------------------------------------------------------------------------
// Kernel 4: out[i] = exp(score[i] - gmax) in place; per-block partial sums
// ---------------------------------------------------------------------------
__global__ void k_exp_sum(float* __restrict__ out,
                          const float* __restrict__ gmax,
                          float* __restrict__ bsum) {
    __shared__ float ss[256];
    const int i = blockIdx.x * 256 + threadIdx.x;
    float e = expf(out[i] - *gmax);
    out[i] = e;
    ss[threadIdx.x] = e;
    __syncthreads();
    for (int off = 128; off > 0; off >>= 1) {
        if ((int)threadIdx.x < off)
            ss[threadIdx.x] += ss[threadIdx.x + off];
        __syncthreads();
    }
    if (threadIdx.x == 0) bsum[blockIdx.x] = ss[0];
}

// ---------------------------------------------------------------------------
// Kernel 5: reduce 256 partial sums -> global sum (single block, deterministic)
// ---------------------------------------------------------------------------
__global__ void k_reduce_sum(const float* __restrict__ bsum,
                             float* __restrict__ gsum) {
    __shared__ float ss[256];
    ss[threadIdx.x] = bsum[threadIdx.x];
    __syncthreads();
    for (int off = 128; off > 0; off >>= 1) {
        if ((int)threadIdx.x < off)
            ss[threadIdx.x] += ss[threadIdx.x + off];
        __syncthreads();
    }
    if (threadIdx.x == 0) *gsum = ss[0];
}

// ---------------------------------------------------------------------------
// Kernel 6: out[i] /= gsum
// ---------------------------------------------------------------------------
__global__ void k_normalize(float* __restrict__ out,
                            const float* __restrict__ gsum) {
    const int i = blockIdx.x * 256 + threadIdx.x;
    out[i] = out[i] / *gsum;
}

// ---------------------------------------------------------------------------
// Launch: inputs are (hidden[1024], encoder_outputs[65536*1024], W[1024*1024],
// b[1024]) all f32. b is unused: softmax is invariant to the uniform shift
// b.hidden. d_out = attn [65536] f32.
// ---------------------------------------------------------------------------
extern "C" void kernel_launch(void* const* d_in, const int* in_sizes, int n_in,
                              void* d_out, int out_size, void* d_ws, size_t ws_size,
                              hipStream_t stream) {
    const float* hidden = (const float*)d_in[0];
    const float* enc    = (const float*)d_in[1];
    const float* W      = (const float*)d_in[2];
    // d_in[3] (b) intentionally unused: cancels in softmax.

    float* out = (float*)d_out;
    float* ws  = (float*)d_ws;

    k_compute_v  <<<H / 256, 256, 0, stream>>>(W, hidden, ws + WS_V);
    k_scores_wmma<<<S / 128, 256, 0, stream>>>(enc, ws + WS_V, out, ws + WS_BMAX);
    k_reduce_max <<<1,       256, 0, stream>>>(ws + WS_BMAX, ws + WS_GMAX);
    k_exp_sum    <<<S / 256, 256, 0, stream>>>(out, ws + WS_GMAX, ws + WS_BSUM);
    k_reduce_sum <<<1,       256, 0, stream>>>(ws + WS_BSUM, ws + WS_GSUM);
    k_normalize  <<<S / 256, 256, 0, stream>>>(out, ws + WS_GSUM);
}